// InfoNCELoss_48610439856644
// MI455X (gfx1250) — compile-verified
//
#include <hip/hip_runtime.h>
#include <hip/hip_bf16.h>
#include <math.h>

// ---------------------------------------------------------------------------
// InfoNCE loss, MI455X (gfx1250, wave32).
// Bandwidth-bound: 532 MB streamed once, ~0.27 GFLOP (AI ~ 0.5 FLOP/B).
// Dot products on the native f32 matrix pipe (V_WMMA_F32_16X16X4_F32),
// anchor chunk in column 0 of B. Row stream uses b128 loads; anchor is staged
// in LDS and fed to B via ds_load_b128 so VMEM carries only the negatives.
// ---------------------------------------------------------------------------

typedef float v2f __attribute__((ext_vector_type(2)));
typedef float v8f __attribute__((ext_vector_type(8)));

#define D_DIM 256
#define N_NEG 127
#define ROWS  128          // 1 positive + 127 negatives
#define INV_TEMP (1.0f / 0.07f)
#define NORM_EPS 1e-12f

static __device__ __forceinline__ v8f wmma_f32_k4(v2f A, v2f B, v8f C) {
    return __builtin_amdgcn_wmma_f32_16x16x4_f32(
        /*neg_a=*/false, A, /*neg_b=*/false, B,
        /*c_mod=*/(short)0, C, /*reuse_a=*/false, /*reuse_b=*/false);
}

__global__ __launch_bounds__(256) void infonce_main(
    const float* __restrict__ anchor,     // [B, 256]
    const float* __restrict__ positive,   // [B, 256]
    const float* __restrict__ negatives,  // [B, 127, 256]
    float* __restrict__ per_b,            // [B] partial losses (already / B)
    float inv_batch)
{
    const int b    = blockIdx.x;
    const int t    = threadIdx.x;
    const int lane = t & 31;
    const int w    = t >> 5;        // wave id 0..7 -> rows [16w, 16w+16)
    const int mrow = lane & 15;     // M index within the wave's 16-row tile
    const int h    = lane >> 4;     // K-half: lane<16 -> d=k..k+3, lane>=16 -> d=k+4..k+7
    const int row  = 16 * w + mrow; // 0..127

    __shared__ __align__(16) float a_lds[D_DIM];
    __shared__ float scores[ROWS];
    __shared__ float rowSS[ROWS];
    __shared__ float logitArr[ROWS];
    __shared__ float red[256];
    __shared__ float aSS_sh;

    // --- Stage anchor row into LDS once (coalesced), and compute ||a||^2 ---
    const float* aptr = anchor + (size_t)b * D_DIM;
    {
        const float av_t = aptr[t];     // 256 threads, 1 float each
        a_lds[t] = av_t;
        red[t]   = av_t * av_t;
    }
    __syncthreads();
    for (int s = 128; s > 0; s >>= 1) {
        if (t < s) red[t] += red[t + s];
        __syncthreads();
    }
    if (t == 0) aSS_sh = red[0];

    // --- Row pointer: row 0 = positive, rows 1..127 = negatives ---
    const float* vptr = (row == 0)
        ? (positive + (size_t)b * D_DIM)
        : (negatives + ((size_t)b * N_NEG + (size_t)(row - 1)) * (size_t)D_DIM);

    v8f   c   = {0.f, 0.f, 0.f, 0.f, 0.f, 0.f, 0.f, 0.f};
    float ssV = 0.f;                 // row sum-of-squares (this lane's d-half)
    const bool col0 = (mrow == 0);   // only column 0 of B carries the anchor

    // K loop: 32 iterations, one b128 row load + one b128 LDS anchor load,
    // two V_WMMA_F32_16X16X4_F32 per iteration. The K-slot <-> d mapping is
    // the same permutation for A and B, so the accumulated dot is exact.
    #pragma unroll 4
    for (int k = 0; k < D_DIM; k += 8) {
        const int d = k + 4 * h;
        const float4 rv = *(const float4*)(vptr + d);     // global_load_b128
        const float4 av = *(const float4*)(&a_lds[d]);    // ds_load_b128

        v2f A0; A0.x = rv.x; A0.y = rv.y;
        v2f A1; A1.x = rv.z; A1.y = rv.w;
        v2f B0; B0.x = col0 ? av.x : 0.f; B0.y = col0 ? av.y : 0.f;
        v2f B1; B1.x = col0 ? av.z : 0.f; B1.y = col0 ? av.w : 0.f;

        c = wmma_f32_k4(A0, B0, c);
        c = wmma_f32_k4(A1, B1, c);

        ssV += rv.x * rv.x + rv.y * rv.y + rv.z * rv.z + rv.w * rv.w;
    }

    // Merge the two d-halves (lanes L and L+16 cover complementary d's).
    ssV += __shfl_xor(ssV, 16, 32);
    if (lane < 16) rowSS[row] = ssV;

    // Column 0 of D: lane 0 holds M=0..7 in c[0..7], lane 16 holds M=8..15.
    if (col0) {
        const int mbase = 16 * w + 8 * h;
        #pragma unroll
        for (int j = 0; j < 8; ++j) scores[mbase + j] = c[j];
    }
    __syncthreads();

    // logits: (dot / (max(||a||,eps)*max(||v||,eps))) / T
    float logit = -INFINITY;
    if (t < ROWS) {
        const float na = fmaxf(sqrtf(aSS_sh),   NORM_EPS);
        const float nv = fmaxf(sqrtf(rowSS[t]), NORM_EPS);
        logit = (scores[t] / (na * nv)) * INV_TEMP;
        logitArr[t] = logit;
    }
    red[t] = (t < ROWS) ? logit : -INFINITY;
    __syncthreads();

    // deterministic max-reduce over 256 slots (upper 128 are -inf)
    for (int s = 128; s > 0; s >>= 1) {
        if (t < s) red[t] = fmaxf(red[t], red[t + s]);
        __syncthreads();
    }
    const float m = red[0];
    __syncthreads();

    red[t] = (t < ROWS) ? expf(logit - m) : 0.f;
    __syncthreads();
    for (int s = 128; s > 0; s >>= 1) {
        if (t < s) red[t] += red[t + s];
        __syncthreads();
    }

    if (t == 0) {
        const float lse = m + logf(red[0]);
        per_b[b] = (lse - logitArr[0]) * inv_batch;  // label 0 == positive row
    }
}

// Deterministic single-block mean over the per-sample partials.
__global__ __launch_bounds__(256) void infonce_reduce(
    const float* __restrict__ per_b, float* __restrict__ out, int n)
{
    __shared__ float red[256];
    const int t = threadIdx.x;
    float s = 0.f;
    for (int i = t; i < n; i += 256) s += per_b[i];   // fixed order per thread
    red[t] = s;
    __syncthreads();
    for (int k = 128; k > 0; k >>= 1) {
        if (t < k) red[t] += red[t + k];
        __syncthreads();
    }
    if (t == 0) out[0] = red[0];
}

extern "C" void kernel_launch(void* const* d_in, const int* in_sizes, int n_in,
                              void* d_out, int out_size, void* d_ws, size_t ws_size,
                              hipStream_t stream) {
    const float* anchor    = (const float*)d_in[0];   // [B,256] f32
    const float* positive  = (const float*)d_in[1];   // [B,256] f32
    const float* negatives = (const float*)d_in[2];   // [B,127,256] f32
    float*       out       = (float*)d_out;           // scalar loss
    float*       per_b     = (float*)d_ws;            // B floats of scratch

    const int B = in_sizes[0] / D_DIM;                // 4096
    const float inv_batch = 1.0f / (float)B;

    infonce_main<<<B, 256, 0, stream>>>(anchor, positive, negatives, per_b, inv_batch);
    infonce_reduce<<<1, 256, 0, stream>>>(per_b, out, B);
}